// FactorizedDistribution_84310208020519
// MI455X (gfx1250) — compile-verified
//
#include <hip/hip_runtime.h>

// Factorized-distribution Gumbel-max sampler for MI455X (gfx1250).
// Memory-bound streaming kernel: ~352 MB traffic -> ~15 us floor @ 23.3 TB/s.
// No matmul structure -> no WMMA; CDNA5 paths used: b128 NT loads/stores,
// async global->LDS table broadcast (ASYNCcnt), LDS b128 gather.

typedef float f32x4 __attribute__((ext_vector_type(4)));
typedef int   i32x4 __attribute__((ext_vector_type(4)));

#define FD_G 8
#define FD_QTAB (8 + (FD_G - 1) * 64)   // 456 floats: q0 (8) + qg (7*64)
#define FD_CHUNKS (FD_QTAB / 4)         // 114 x 16B chunks

#if defined(__has_builtin)
#if __has_builtin(__builtin_amdgcn_global_load_async_to_lds_b128)
#define FD_ASYNC_BUILTIN 1
#endif
#endif

// Builtin signature (from hipcc diagnostic): (v4i* AS1, v4i* AS3, imm, imm)
typedef __attribute__((address_space(1))) i32x4* fd_gp;
typedef __attribute__((address_space(3))) i32x4* fd_lp;

__device__ __forceinline__ void fd_wait_async0(void) {
#if defined(__has_builtin) && __has_builtin(__builtin_amdgcn_s_wait_asynccnt)
  __builtin_amdgcn_s_wait_asynccnt(0);
#else
  asm volatile("s_wait_asynccnt 0" ::: "memory");
#endif
}

// argmax over 8 candidates q[i] + n[i]; strict '>' keeps first max (jnp.argmax ties)
__device__ __forceinline__ int fd_argmax8(f32x4 qa, f32x4 qb, f32x4 na, f32x4 nb) {
  float v0 = qa.x + na.x, v1 = qa.y + na.y, v2 = qa.z + na.z, v3 = qa.w + na.w;
  float v4 = qb.x + nb.x, v5 = qb.y + nb.y, v6 = qb.z + nb.z, v7 = qb.w + nb.w;
  int best = 0; float bv = v0;
  if (v1 > bv) { bv = v1; best = 1; }
  if (v2 > bv) { bv = v2; best = 2; }
  if (v3 > bv) { bv = v3; best = 3; }
  if (v4 > bv) { bv = v4; best = 4; }
  if (v5 > bv) { bv = v5; best = 5; }
  if (v6 > bv) { bv = v6; best = 6; }
  if (v7 > bv) { bv = v7; best = 7; }
  return best;
}

__global__ __launch_bounds__(256) void
fd_sample_kernel(const float* __restrict__ q0,
                 const float* __restrict__ qg,
                 const float* __restrict__ noise0,
                 const float* __restrict__ noise,
                 float* __restrict__ out,
                 int n_samples) {
  __shared__ float qtab[FD_QTAB];
  const int tid = threadIdx.x;

  // --- Broadcast the 1.8 KB q-table into LDS via the CDNA5 async path ---
  if (tid < FD_CHUNKS) {
    const float* src = (tid < 2) ? (q0 + tid * 4) : (qg + (tid - 2) * 4);
#if FD_ASYNC_BUILTIN
    __builtin_amdgcn_global_load_async_to_lds_b128(
        (fd_gp)src, (fd_lp)&qtab[tid * 4], /*offset=*/0, /*cpol=*/0);
#else
    unsigned lds_off = (unsigned)(size_t)(fd_lp)&qtab[tid * 4];
    asm volatile("global_load_async_to_lds_b128 %0, %1, off"
                 :: "v"(lds_off), "v"((unsigned long long)(size_t)src)
                 : "memory");
#endif
  }
  fd_wait_async0();
  __syncthreads();

  const int n = (int)(blockIdx.x * 256u + (unsigned)tid);
  if (n >= n_samples) return;

  // --- Issue all 16 b128 noise loads up front (addresses independent of the
  //     sequential argmax chain) as non-temporal streaming reads ---
  f32x4 nl[FD_G][2];
  {
    const f32x4* p0 = (const f32x4*)(noise0 + (size_t)n * 8);
    nl[0][0] = __builtin_nontemporal_load(p0);
    nl[0][1] = __builtin_nontemporal_load(p0 + 1);
#pragma unroll
    for (int g = 1; g < FD_G; ++g) {
      const f32x4* pg =
          (const f32x4*)(noise + ((size_t)(g - 1) * (size_t)n_samples + (size_t)n) * 8);
      nl[g][0] = __builtin_nontemporal_load(pg);
      nl[g][1] = __builtin_nontemporal_load(pg + 1);
    }
  }

  float bits[FD_G * 3];
  int prev;
  {
    const f32x4* qp = (const f32x4*)qtab;       // group 0: unconditional table
    prev = fd_argmax8(qp[0], qp[1], nl[0][0], nl[0][1]);
    bits[0] = (float)((prev >> 2) & 1);
    bits[1] = (float)((prev >> 1) & 1);
    bits[2] = (float)(prev & 1);
  }
#pragma unroll
  for (int g = 1; g < FD_G; ++g) {
    // per-lane row gather: row index = previous group's 3 bits (== prev)
    // LDS layout puts row r of group g at float offset 8 + 64*(g-1) + 8*r
    // -> 16B-aligned, rows map to disjoint bank octets (conflict-free)
    const f32x4* qp = (const f32x4*)(qtab + 8 + (g - 1) * 64 + prev * 8);
    prev = fd_argmax8(qp[0], qp[1], nl[g][0], nl[g][1]);
    bits[g * 3 + 0] = (float)((prev >> 2) & 1);
    bits[g * 3 + 1] = (float)((prev >> 1) & 1);
    bits[g * 3 + 2] = (float)(prev & 1);
  }

  // --- 24 floats per sample, 16B-aligned (n*96 bytes) -> 6 NT b128 stores ---
  f32x4* op = (f32x4*)(out + (size_t)n * 24);
#pragma unroll
  for (int k = 0; k < 6; ++k) {
    f32x4 o = { bits[4 * k + 0], bits[4 * k + 1], bits[4 * k + 2], bits[4 * k + 3] };
    __builtin_nontemporal_store(o, op + k);
  }
}

extern "C" void kernel_launch(void* const* d_in, const int* in_sizes, int n_in,
                              void* d_out, int out_size, void* d_ws, size_t ws_size,
                              hipStream_t stream) {
  const float* q0     = (const float*)d_in[0];  // (2,2,2)            = 8
  const float* qg     = (const float*)d_in[1];  // (7,2,2,2,2,2,2)    = 448
  const float* noise0 = (const float*)d_in[2];  // (N,2,2,2)          = N*8
  const float* noise  = (const float*)d_in[3];  // (7,N,2,2,2)        = 7*N*8
  float* out = (float*)d_out;                   // (N, 24)

  const int n_samples = in_sizes[2] / 8;
  const int blocks = (n_samples + 255) / 256;
  fd_sample_kernel<<<blocks, 256, 0, stream>>>(q0, qg, noise0, noise, out, n_samples);

  (void)n_in; (void)out_size; (void)d_ws; (void)ws_size;
}